// _IGCL_26929444946277
// MI455X (gfx1250) — compile-verified
//
#include <hip/hip_runtime.h>

typedef __attribute__((ext_vector_type(2))) float v2f;
typedef __attribute__((ext_vector_type(8))) float v8f;

#define NUSER 100000
#define NITEM 50000
#define NNODE 150000
#define DIM   64
#define RDIM  32
#define NELEM (NNODE * DIM)   // 9,600,000 floats
#define WPB   4               // waves per block in the WMMA kernel

// ---------------- elementwise helpers ----------------

__global__ void fill_zero(float* __restrict__ p, int n) {
  int i = blockIdx.x * blockDim.x + threadIdx.x;
  if (i < n) p[i] = 0.0f;
}

__global__ void init_concat(const float* __restrict__ ue, const float* __restrict__ ie,
                            float* __restrict__ acc, float* __restrict__ cur,
                            int nuser_elems, int n) {
  int i = blockIdx.x * blockDim.x + threadIdx.x;
  if (i >= n) return;
  float v = (i < nuser_elems) ? ue[i] : ie[i - nuser_elems];
  acc[i] = v;
  cur[i] = v;
}

__global__ void acc_add(float* __restrict__ acc, const float* __restrict__ nxt, int n) {
  int i = blockIdx.x * blockDim.x + threadIdx.x;
  if (i < n) acc[i] += nxt[i];
}

__global__ void zero_loss(float* __restrict__ out) { out[NELEM] = 0.0f; }

// ---------------- SpMM: y += A * x (edge-parallel atomic scatter) ----------------
// 16 threads per edge; each handles a float4 chunk of the 64-wide row.
// x (38.4 MB) and the adjacency (57.6 MB) both fit in the 192 MB L2, so the
// gathers and atomics resolve at L2 bandwidth.

__global__ void spmm_scatter(const int* __restrict__ rows, const int* __restrict__ cols,
                             const float* __restrict__ vals,
                             const float* __restrict__ x, float* __restrict__ y,
                             long long total) {
  long long t = (long long)blockIdx.x * blockDim.x + threadIdx.x;
  if (t >= total) return;
  int e = (int)(t >> 4);
  int j = (int)(t & 15) * 4;
  int r = rows[e];
  int c = cols[e];
  float v = vals[e];
  const float4 xv = *(const float4*)(x + (size_t)c * DIM + j);
  float* yp = y + (size_t)r * DIM + j;
  atomicAdd(yp + 0, v * xv.x);
  atomicAdd(yp + 1, v * xv.y);
  atomicAdd(yp + 2, v * xv.z);
  atomicAdd(yp + 3, v * xv.w);
}

// ---------------- fused MLP + loss via V_WMMA_F32_16X16X4_F32 ----------------
// One wave per 16-row tile. X tile staged in LDS (scaled by 1/4). fc1 produces
// the 16x32 hidden via 2 accumulators x 16 K-steps of f32 WMMA; hidden goes
// through LDS to re-fragment as an A operand; fc2 produces 4 16x16 output
// tiles. Bias, store, and squared-error loss fused into the epilogue.

__global__ __launch_bounds__(WPB * 32) void mlp_loss(
    const float* __restrict__ acc,
    const float* __restrict__ w1, const float* __restrict__ b1,
    const float* __restrict__ w2, const float* __restrict__ b2,
    float* __restrict__ out, int ngroups) {
  __shared__ float Xs[WPB][16 * DIM];
  __shared__ float Hs[WPB][16 * RDIM];

  const int lane  = threadIdx.x & 31;
  const int wave  = threadIdx.x >> 5;
  const int group = blockIdx.x * WPB + wave;
  if (group >= ngroups) return;          // uniform per-wave: EXEC stays all-1s
  const int row0 = group * 16;

  float* X = Xs[wave];
  float* H = Hs[wave];
  const int l16  = lane & 15;
  const int half = lane >> 4;            // K/M split across lane halves

  // stage 16x64 tile of all_emb = acc/4 (coalesced, 1 global read per element)
  const float* src = acc + (size_t)row0 * DIM;
  #pragma unroll
  for (int i = 0; i < 32; ++i) X[lane + i * 32] = src[lane + i * 32] * 0.25f;

  // fc1: H(16x32) = X(16x64) @ w1^T + b1   — w1 is (32,64) row-major, so
  // B[k][n] = w1[n][k] → contiguous float2 along each w1 row.
  v8f h0 = {}; v8f h1 = {};
  #pragma unroll
  for (int kk = 0; kk < DIM; kk += 4) {
    v2f a  = *(const v2f*)&X[l16 * DIM + kk + half * 2];
    v2f bA = *(const v2f*)&w1[(0  + l16) * DIM + kk + half * 2];
    v2f bB = *(const v2f*)&w1[(16 + l16) * DIM + kk + half * 2];
    h0 = __builtin_amdgcn_wmma_f32_16x16x4_f32(false, a, false, bA, (short)0, h0, false, false);
    h1 = __builtin_amdgcn_wmma_f32_16x16x4_f32(false, a, false, bB, (short)0, h1, false, false);
  }
  const float bias0 = b1[l16];
  const float bias1 = b1[16 + l16];
  #pragma unroll
  for (int r = 0; r < 8; ++r) {
    int M = r + half * 8;                // C/D layout: VGPR r holds M=r, M=r+8
    H[M * RDIM + l16]      = h0[r] + bias0;
    H[M * RDIM + 16 + l16] = h1[r] + bias1;
  }
  // intra-wave LDS RAW: DS ops from one wave execute in order.

  // fc2: G(16x64) = H(16x32) @ w2^T + b2   — w2 is (64,32) row-major.
  float lsum = 0.0f;
  #pragma unroll
  for (int n0 = 0; n0 < DIM; n0 += 16) {
    v8f g = {};
    #pragma unroll
    for (int kk = 0; kk < RDIM; kk += 4) {
      v2f a = *(const v2f*)&H[l16 * RDIM + kk + half * 2];
      v2f b = *(const v2f*)&w2[(n0 + l16) * RDIM + kk + half * 2];
      g = __builtin_amdgcn_wmma_f32_16x16x4_f32(false, a, false, b, (short)0, g, false, false);
    }
    const float bias = b2[n0 + l16];
    #pragma unroll
    for (int r = 0; r < 8; ++r) {
      int M = r + half * 8;
      float gv  = g[r] + bias;
      int   col = n0 + l16;
      out[(size_t)(row0 + M) * DIM + col] = gv;   // u_gen/i_gen are contiguous
      float d = gv - X[M * DIM + col];
      lsum += d * d;
    }
  }
  atomicAdd(out + NELEM, lsum);
}

// ---------------- driver ----------------

extern "C" void kernel_launch(void* const* d_in, const int* in_sizes, int n_in,
                              void* d_out, int out_size, void* d_ws, size_t ws_size,
                              hipStream_t stream) {
  const int*   rows = (const int*)d_in[0];
  const int*   cols = (const int*)d_in[1];
  const float* vals = (const float*)d_in[2];
  const float* ue   = (const float*)d_in[3];
  const float* ie   = (const float*)d_in[4];
  const float* w1   = (const float*)d_in[5];
  const float* b1   = (const float*)d_in[6];
  const float* w2   = (const float*)d_in[7];
  const float* b2   = (const float*)d_in[8];
  float* out = (float*)d_out;
  const int nnz = in_sizes[0];

  float* acc  = (float*)d_ws;      // NELEM floats
  float* bufA = acc + NELEM;       // NELEM floats
  float* bufB = out;               // reuse d_out as ping-pong; overwritten by mlp_loss

  const int  n    = NELEM;
  const int  gb_n = (n + 255) / 256;
  const long long total = (long long)nnz * 16;
  const int  gb_e = (int)((total + 255) / 256);

  init_concat<<<gb_n, 256, 0, stream>>>(ue, ie, acc, bufA, NUSER * DIM, n);

  float* cur = bufA;
  float* nxt = bufB;
  for (int layer = 0; layer < 3; ++layer) {
    fill_zero<<<gb_n, 256, 0, stream>>>(nxt, n);
    spmm_scatter<<<gb_e, 256, 0, stream>>>(rows, cols, vals, cur, nxt, total);
    acc_add<<<gb_n, 256, 0, stream>>>(acc, nxt, n);
    float* t = cur; cur = nxt; nxt = t;
  }

  zero_loss<<<1, 1, 0, stream>>>(out);
  const int ngroups = NNODE / 16;   // 9375
  mlp_loss<<<(ngroups + WPB - 1) / WPB, WPB * 32, 0, stream>>>(acc, w1, b1, w2, b2, out, ngroups);
}